// InferModel_88252987998604
// MI455X (gfx1250) — compile-verified
//
#include <hip/hip_runtime.h>

typedef __attribute__((ext_vector_type(2))) float v2f;
typedef __attribute__((ext_vector_type(8))) float v8f;

#define S_NUM 4
#define E_NUM 8
#define D_IN 1008
#define H0 256
#define H1 192
#define H2 160
#define TM 32          // atoms per tile (2 WMMA row-blocks of 16)
#define NWAVES 8
#define NTHREADS 256
#define CHUNK 336      // layer-0 K chunk (1008 = 3*336; 336/4 = 84 float4 per row)
#define SA 340         // LDS stride for aev chunk (floats, 16B aligned)
#define SH 260         // LDS stride for hidden buffers (floats, 16B aligned)
#define CELU_A 0.1f

#define USE_ASYNC_LDS 1

// ---------------- bucketing kernels ----------------
// ws int layout: [0..3] counts, [4..8] offsets, [9..12] cursors, [16..16+N) perm
// then (256-int aligned) packed W0 / W1 / W2 as floats.

__global__ void init_kernel(int* ws, float* out) {
  int t = threadIdx.x;
  if (t < 16) ws[t] = 0;
  if (t == 0) out[0] = 0.0f;
}

__global__ void count_kernel(const int* __restrict__ species, int n, int* __restrict__ ws) {
  int i = blockIdx.x * blockDim.x + threadIdx.x;
  if (i < n) atomicAdd(&ws[species[i] & 3], 1);
}

__global__ void scan_kernel(int* ws) {
  if (threadIdx.x == 0) {
    int acc = 0;
    for (int s = 0; s < S_NUM; ++s) { ws[4 + s] = acc; acc += ws[s]; }
    ws[8] = acc;
  }
}

__global__ void scatter_kernel(const int* __restrict__ species, int n, int* __restrict__ ws) {
  int i = blockIdx.x * blockDim.x + threadIdx.x;
  if (i < n) {
    int s = species[i] & 3;
    int pos = ws[4 + s] + atomicAdd(&ws[9 + s], 1);
    ws[16 + pos] = i;
  }
}

// Repack W [SE][din][dout] -> [SE][din/4][dout][4]: one lane's B-fragment is a
// single aligned b64; a wave's 32 lanes cover one contiguous 256B segment.
__global__ void repack_kernel(const float* __restrict__ src, float* __restrict__ dst,
                              int din, int dout, int total) {
  int i = blockIdx.x * blockDim.x + threadIdx.x;
  if (i >= total) return;
  const int per = din * dout;
  const int se = i / per;
  const int rem = i - se * per;
  const int k = rem / dout;
  const int n = rem - k * dout;
  dst[(size_t)se * per + (((size_t)(k >> 2) * dout + n) << 2) + (k & 3)] = src[i];
}

// ---------------- fused MLP kernel ----------------

__device__ __forceinline__ float celu_f(float x) {
  return x > 0.0f ? x : CELU_A * (__expf(x * (1.0f / CELU_A)) - 1.0f);
}

// 32-row GEMM: lout[32 x dout] = celu(lin[32 x din] * W + bias), W pre-packed.
// Two row-blocks share each B-fragment (register reuse).
__device__ __forceinline__ void layer_wmma(const float* lin, int sin, int din,
                                           const float* __restrict__ Wp,
                                           const float* __restrict__ bias, int dout,
                                           float* lout, int sout,
                                           int wave, int half, int lr) {
  for (int t = wave; t < (dout >> 4); t += NWAVES) {
    const int n0 = t << 4;
    v8f acc0 = {}, acc1 = {};
    const float* arow0 = lin + lr * sin + 2 * half;
    const float* arow1 = lin + (16 + lr) * sin + 2 * half;
    const float* wcol = Wp + (((size_t)(n0 + lr)) << 2) + 2 * half;
    for (int k0 = 0; k0 < din; k0 += 4) {
      v2f a0, a1, b;
      a0.x = arow0[k0];
      a0.y = arow0[k0 + 1];
      a1.x = arow1[k0];
      a1.y = arow1[k0 + 1];
      b = *(const v2f*)(wcol + (size_t)k0 * dout);
      acc0 = __builtin_amdgcn_wmma_f32_16x16x4_f32(false, a0, false, b, (short)0, acc0,
                                                   false, false);
      acc1 = __builtin_amdgcn_wmma_f32_16x16x4_f32(false, a1, false, b, (short)0, acc1,
                                                   false, false);
    }
    const float bv = bias[n0 + lr];
#pragma unroll
    for (int r = 0; r < 8; ++r) {
      lout[(r + 8 * half) * sout + n0 + lr] = celu_f(acc0[r] + bv);
      lout[(16 + r + 8 * half) * sout + n0 + lr] = celu_f(acc1[r] + bv);
    }
  }
}

__global__ void __launch_bounds__(NTHREADS)
mlp_kernel(const float* __restrict__ aev,
           const float* __restrict__ W0p, const float* __restrict__ b0,
           const float* __restrict__ W1p, const float* __restrict__ b1,
           const float* __restrict__ W2p, const float* __restrict__ b2,
           const float* __restrict__ W3, const float* __restrict__ b3,
           const int* __restrict__ ws, float* __restrict__ out) {
  extern __shared__ float smem[];
  float* ldsA = smem;                 // TM * SA  (aev chunk)
  float* hidA = ldsA + TM * SA;       // TM * SH  (layer0 out / layer2 out)
  float* hidB = hidA + TM * SH;       // TM * SH  (layer1 out)
  float* red = hidB + TM * SH;        // TM

  const int s = blockIdx.y;
  const int tile = blockIdx.x;
  const int base = ws[4 + s];
  const int cnt = ws[5 + s] - base;
  if (tile * TM >= cnt) return;  // uniform early-exit
  const int rows = (cnt - tile * TM) < TM ? (cnt - tile * TM) : TM;

  const int tid = threadIdx.x;
  const int wave = tid >> 5;
  const int lane = tid & 31;
  const int half = lane >> 4;
  const int lr = lane & 15;
  const int* perm = ws + 16 + base + tile * TM;

  float e_acc = 0.0f;
  for (int e = 0; e < E_NUM; ++e) {
    const int se = s * E_NUM + e;
    const float* W0e = W0p + (size_t)se * D_IN * H0;
    const float* b0e = b0 + se * H0;

    // ---- layer 0: 1008 -> 256, A staged in LDS chunks of 336 cols ----
    // wave handles col tiles {wave, wave+8} x row blocks {0,1}
    const int n0a = wave << 4;
    const int n0b = (wave + 8) << 4;
    v8f acc00 = {}, acc01 = {}, acc10 = {}, acc11 = {};
    for (int ch = 0; ch < D_IN / CHUNK; ++ch) {
      __syncthreads();  // previous chunk fully consumed by all waves
      for (int idx = tid; idx < TM * (CHUNK / 4); idx += NTHREADS) {
        const int row = idx / (CHUNK / 4);
        const int c4 = idx - row * (CHUNK / 4);
        const int r = row < rows ? row : rows - 1;  // pad by replicating last row
        const float* g = aev + (size_t)perm[r] * D_IN + ch * CHUNK + c4 * 4;
#if USE_ASYNC_LDS
        unsigned lds_byte = (unsigned)(uintptr_t)(const void*)(ldsA + row * SA + c4 * 4);
        unsigned long long ga = (unsigned long long)(uintptr_t)(const void*)g;
        asm volatile("global_load_async_to_lds_b128 %0, %1, off"
                     :: "v"(lds_byte), "v"(ga)
                     : "memory");
#else
        const float4 v = *(const float4*)g;
        *(float4*)(ldsA + row * SA + c4 * 4) = v;
#endif
      }
#if USE_ASYNC_LDS
      asm volatile("s_wait_asynccnt 0" ::: "memory");
#endif
      __syncthreads();
      const float* arow0 = ldsA + lr * SA + 2 * half;
      const float* arow1 = ldsA + (16 + lr) * SA + 2 * half;
      const float* wbase = W0e + (size_t)(ch * CHUNK) * H0 + 2 * half;
      for (int k0 = 0; k0 < CHUNK; k0 += 4) {
        v2f a0, a1, b0v, b1v;
        a0.x = arow0[k0];
        a0.y = arow0[k0 + 1];
        a1.x = arow1[k0];
        a1.y = arow1[k0 + 1];
        b0v = *(const v2f*)(wbase + (size_t)k0 * H0 + ((n0a + lr) << 2));
        b1v = *(const v2f*)(wbase + (size_t)k0 * H0 + ((n0b + lr) << 2));
        acc00 = __builtin_amdgcn_wmma_f32_16x16x4_f32(false, a0, false, b0v, (short)0,
                                                      acc00, false, false);
        acc10 = __builtin_amdgcn_wmma_f32_16x16x4_f32(false, a1, false, b0v, (short)0,
                                                      acc10, false, false);
        acc01 = __builtin_amdgcn_wmma_f32_16x16x4_f32(false, a0, false, b1v, (short)0,
                                                      acc01, false, false);
        acc11 = __builtin_amdgcn_wmma_f32_16x16x4_f32(false, a1, false, b1v, (short)0,
                                                      acc11, false, false);
      }
    }
    {
      const float bva = b0e[n0a + lr];
      const float bvb = b0e[n0b + lr];
#pragma unroll
      for (int r = 0; r < 8; ++r) {
        hidA[(r + 8 * half) * SH + n0a + lr] = celu_f(acc00[r] + bva);
        hidA[(16 + r + 8 * half) * SH + n0a + lr] = celu_f(acc10[r] + bva);
        hidA[(r + 8 * half) * SH + n0b + lr] = celu_f(acc01[r] + bvb);
        hidA[(16 + r + 8 * half) * SH + n0b + lr] = celu_f(acc11[r] + bvb);
      }
    }
    __syncthreads();

    // ---- layer 1: 256 -> 192 ----
    layer_wmma(hidA, SH, H0, W1p + (size_t)se * H0 * H1, b1 + se * H1, H1, hidB, SH,
               wave, half, lr);
    __syncthreads();

    // ---- layer 2: 192 -> 160 ----
    layer_wmma(hidB, SH, H1, W2p + (size_t)se * H1 * H2, b2 + se * H2, H2, hidA, SH,
               wave, half, lr);
    __syncthreads();

    // ---- layer 3: 160 -> 1 (per-atom dot) ----
    if (tid < TM) {
      const float* h = hidA + tid * SH;
      const float* w3 = W3 + (size_t)se * H2;
      float sum = 0.0f;
      for (int k = 0; k < H2; ++k) sum += h[k] * w3[k];
      e_acc += sum + b3[se];
    }
    __syncthreads();
  }

  if (tid < TM) red[tid] = (tid < rows) ? e_acc * (1.0f / E_NUM) : 0.0f;
  __syncthreads();
  if (tid == 0) {
    float t = 0.0f;
    for (int i = 0; i < TM; ++i) t += red[i];
    atomicAdd(out, t);
  }
}

// ---------------- host launcher ----------------

extern "C" void kernel_launch(void* const* d_in, const int* in_sizes, int n_in,
                              void* d_out, int out_size, void* d_ws, size_t ws_size,
                              hipStream_t stream) {
  (void)n_in; (void)out_size; (void)ws_size;
  const float* aev = (const float*)d_in[0];
  const int* species = (const int*)d_in[1];
  const float* W0 = (const float*)d_in[2];
  const float* b0 = (const float*)d_in[3];
  const float* W1 = (const float*)d_in[4];
  const float* b1 = (const float*)d_in[5];
  const float* W2 = (const float*)d_in[6];
  const float* b2 = (const float*)d_in[7];
  const float* W3 = (const float*)d_in[8];
  const float* b3 = (const float*)d_in[9];
  float* out = (float*)d_out;
  int* ws = (int*)d_ws;
  float* wsf = (float*)d_ws;
  const int N = in_sizes[1];

  const size_t SE = (size_t)S_NUM * E_NUM;
  const size_t pw0 = ((size_t)(16 + N) + 255) & ~(size_t)255;
  const size_t pw1 = pw0 + SE * D_IN * H0;
  const size_t pw2 = pw1 + SE * H0 * H1;

  init_kernel<<<1, 32, 0, stream>>>(ws, out);
  count_kernel<<<(N + 255) / 256, 256, 0, stream>>>(species, N, ws);
  scan_kernel<<<1, 1, 0, stream>>>(ws);
  scatter_kernel<<<(N + 255) / 256, 256, 0, stream>>>(species, N, ws);

  const int t0 = (int)(SE * D_IN * H0);
  const int t1 = (int)(SE * H0 * H1);
  const int t2 = (int)(SE * H1 * H2);
  repack_kernel<<<(t0 + 255) / 256, 256, 0, stream>>>(W0, wsf + pw0, D_IN, H0, t0);
  repack_kernel<<<(t1 + 255) / 256, 256, 0, stream>>>(W1, wsf + pw1, H0, H1, t1);
  repack_kernel<<<(t2 + 255) / 256, 256, 0, stream>>>(W2, wsf + pw2, H1, H2, t2);

  const size_t lds_bytes = (size_t)(TM * SA + 2 * TM * SH + TM) * sizeof(float);
  dim3 grid((N + TM - 1) / TM, S_NUM);
  mlp_kernel<<<grid, NTHREADS, lds_bytes, stream>>>(aev, wsf + pw0, b0, wsf + pw1, b1,
                                                    wsf + pw2, b2, W3, b3, ws, out);
}